// GCNEEGNet_77335181132254
// MI455X (gfx1250) — compile-verified
//
#include <hip/hip_runtime.h>
#include <hip/hip_bf16.h>

// ---- problem constants (from reference) ----
constexpr int Bd = 32;
constexpr int Cd = 64;
constexpr int Td = 4000;
constexpr int F1d = 8;
constexpr int F2d = 16;
constexpr int CH16 = 16;   // F1*D

typedef __attribute__((ext_vector_type(2))) float v2f;
typedef __attribute__((ext_vector_type(8))) float v8f;

// -------------------------------------------------------------------------
// 0) Aw[f,c,d] = adjacency[c,d] * imp[l,f,c,d]   (F1*64*64 = 32768 floats)
__global__ void k_prep_A(const float* __restrict__ adj, const float* __restrict__ impl,
                         float* __restrict__ aw) {
    int i = blockIdx.x * 256 + threadIdx.x;          // 0..32767
    aw[i] = adj[i & (Cd * Cd - 1)] * impl[i];
}

// -------------------------------------------------------------------------
// 1) conv1 (1 -> F1, K=64 along T, pad 31/32) + BN(bnF1[0])
// grid: (16 t-tiles, C, B), block 256
__global__ void k_conv1_bn(const float* __restrict__ x, const float* __restrict__ w,
                           const float* __restrict__ bn, float* __restrict__ out) {
    __shared__ float ws[F1d * 64];
    __shared__ float xs[320];
    __shared__ float sc[F1d], sh[F1d];
    int tid = threadIdx.x;
    int t0 = blockIdx.x * 256;
    int c = blockIdx.y, b = blockIdx.z;
    for (int i = tid; i < F1d * 64; i += 256) ws[i] = w[i];
    for (int i = tid; i < 319; i += 256) {
        int xi = t0 - 31 + i;
        xs[i] = (xi >= 0 && xi < Td) ? x[((size_t)b * Cd + c) * Td + xi] : 0.f;
    }
    if (tid < F1d) {
        float g = bn[0 * F1d + tid], bb = bn[1 * F1d + tid];
        float m = bn[2 * F1d + tid], v = bn[3 * F1d + tid];
        float s = g * rsqrtf(v + 1e-3f);
        sc[tid] = s; sh[tid] = bb - m * s;
    }
    __syncthreads();
    int t = t0 + tid;
    if (t < Td) {
        for (int f = 0; f < F1d; ++f) {
            float acc = 0.f;
            #pragma unroll 8
            for (int k = 0; k < 64; ++k) acc += ws[f * 64 + k] * xs[tid + k];
            out[(((size_t)b * F1d + f) * Cd + c) * Td + t] = acc * sc[f] + sh[f];
        }
    }
}

// -------------------------------------------------------------------------
// 2) einsum('fcd,bfdt->bfct'): per (b,f) a 64x64 @ 64x4000 f32 matmul.
// One wave computes a full 64-row x 16-col output block (4 WMMA acc chains),
// so B (the activation) is loaded exactly once per layer.
// block (32,2) = 2 waves; grid (125 t-tile pairs, B*F1)
__global__ void k_einsum_wmma(const float* __restrict__ src, float* __restrict__ dst,
                              const float* __restrict__ Aw) {
    int lane = threadIdx.x;                    // 0..31
    int tt = blockIdx.x * 2 + threadIdx.y;     // 0..249  (t tile)
    int bf = blockIdx.y;                       // 0..255  (b*F1+f)
    int f = bf & (F1d - 1);
    const float* hs = src + (size_t)bf * (Cd * Td);
    float*       hd = dst + (size_t)bf * (Cd * Td);
    const float* Af = Aw + (size_t)f * Cd * Cd;

    int M = lane & 15;                 // A row within tile / B-D column within tile
    int kh = (lane >> 4) << 1;         // half-wave K offset (0 or 2)
    int ncol = tt * 16 + M;            // global N column (t index)

    v8f acc0 = {}, acc1 = {}, acc2 = {}, acc3 = {};
    for (int kb = 0; kb < 16; ++kb) {  // K = 64, 4 per WMMA
        int k0 = kb * 4 + kh;
        v2f bv;
        bv.x = hs[(size_t)k0 * Td + ncol];
        bv.y = hs[(size_t)(k0 + 1) * Td + ncol];
        v2f a0 = *(const v2f*)(Af + (size_t)(0 * 16 + M) * Cd + k0);
        v2f a1 = *(const v2f*)(Af + (size_t)(1 * 16 + M) * Cd + k0);
        v2f a2 = *(const v2f*)(Af + (size_t)(2 * 16 + M) * Cd + k0);
        v2f a3 = *(const v2f*)(Af + (size_t)(3 * 16 + M) * Cd + k0);
        acc0 = __builtin_amdgcn_wmma_f32_16x16x4_f32(false, a0, false, bv, (short)0, acc0, false, false);
        acc1 = __builtin_amdgcn_wmma_f32_16x16x4_f32(false, a1, false, bv, (short)0, acc1, false, false);
        acc2 = __builtin_amdgcn_wmma_f32_16x16x4_f32(false, a2, false, bv, (short)0, acc2, false, false);
        acc3 = __builtin_amdgcn_wmma_f32_16x16x4_f32(false, a3, false, bv, (short)0, acc3, false, false);
    }
    int rhalf = (lane >> 4) << 3;      // lanes 16..31 hold rows M+8 of each tile
    #pragma unroll
    for (int r = 0; r < 8; ++r) {
        hd[(size_t)( 0 + rhalf + r) * Td + ncol] = acc0[r];
        hd[(size_t)(16 + rhalf + r) * Td + ncol] = acc1[r];
        hd[(size_t)(32 + rhalf + r) * Td + ncol] = acc2[r];
        hd[(size_t)(48 + rhalf + r) * Td + ncol] = acc3[r];
    }
}

// -------------------------------------------------------------------------
// 3) gconv (F1->F1, K=8 along T, pad 3/4) + BN(bnF1[l+1]) + ELU
// grid: (16 t-tiles, C, B), block 256
__global__ void k_gconv_bn_elu(const float* __restrict__ src, float* __restrict__ dst,
                               const float* __restrict__ gwl, const float* __restrict__ bn) {
    __shared__ float wsh[F1d * F1d * 8];
    __shared__ float xs[F1d][264];
    __shared__ float sc[F1d], sh[F1d];
    int tid = threadIdx.x;
    int t0 = blockIdx.x * 256;
    int c = blockIdx.y, b = blockIdx.z;
    for (int i = tid; i < F1d * F1d * 8; i += 256) wsh[i] = gwl[i];
    for (int i = tid; i < F1d * 263; i += 256) {
        int f2 = i / 263, j = i % 263;
        int xi = t0 - 3 + j;
        xs[f2][j] = (xi >= 0 && xi < Td)
                        ? src[(((size_t)b * F1d + f2) * Cd + c) * Td + xi] : 0.f;
    }
    if (tid < F1d) {
        float g = bn[0 * F1d + tid], bb = bn[1 * F1d + tid];
        float m = bn[2 * F1d + tid], v = bn[3 * F1d + tid];
        float s = g * rsqrtf(v + 1e-3f);
        sc[tid] = s; sh[tid] = bb - m * s;
    }
    __syncthreads();
    int t = t0 + tid;
    if (t < Td) {
        for (int f = 0; f < F1d; ++f) {
            float acc = 0.f;
            for (int f2 = 0; f2 < F1d; ++f2) {
                #pragma unroll
                for (int k = 0; k < 8; ++k)
                    acc += wsh[(f * F1d + f2) * 8 + k] * xs[f2][tid + k];
            }
            float y = acc * sc[f] + sh[f];
            y = (y > 0.f) ? y : (__expf(y) - 1.f);
            dst[(((size_t)b * F1d + f) * Cd + c) * Td + t] = y;
        }
    }
}

// -------------------------------------------------------------------------
// 4) max-norm depthwise conv over C (groups=F1, 2 out/group) + BN(bn_sp)
//    + ELU + avg-pool 4 along T.  p: (B,16,1000)
// grid: (4 t4-tiles of 250, 16 channels, B), block 256
__global__ void k_dw_bn_elu_pool4(const float* __restrict__ h, const float* __restrict__ dww,
                                  const float* __restrict__ bnsp, float* __restrict__ p) {
    __shared__ float wr[Cd];
    __shared__ float sb[3];
    int tid = threadIdx.x;
    int m = blockIdx.y, b = blockIdx.z;
    int g = m >> 1;
    if (tid < Cd) wr[tid] = dww[m * Cd + tid];
    __syncthreads();
    if (tid == 0) {
        float n2 = 0.f;
        for (int c = 0; c < Cd; ++c) n2 += wr[c] * wr[c];
        float s = fminf(1.f, 1.f / fmaxf(sqrtf(n2), 1e-7f));
        sb[0] = s;
        float gg = bnsp[0 * CH16 + m], bb = bnsp[1 * CH16 + m];
        float mm = bnsp[2 * CH16 + m], vv = bnsp[3 * CH16 + m];
        float bs = gg * rsqrtf(vv + 1e-3f);
        sb[1] = bs; sb[2] = bb - mm * bs;
    }
    __syncthreads();
    if (tid < 250) {
        int t4 = blockIdx.x * 250 + tid;
        float a0 = 0, a1 = 0, a2 = 0, a3 = 0;
        float wsc = sb[0];
        const float* base = h + ((size_t)(b * F1d + g) * Cd) * Td + (size_t)t4 * 4;
        for (int c = 0; c < Cd; ++c) {
            float4 xv = *(const float4*)(base + (size_t)c * Td);
            float w = wr[c] * wsc;
            a0 += w * xv.x; a1 += w * xv.y; a2 += w * xv.z; a3 += w * xv.w;
        }
        float bs = sb[1], bsh = sb[2], s = 0.f, y;
        y = a0 * bs + bsh; s += (y > 0.f) ? y : (__expf(y) - 1.f);
        y = a1 * bs + bsh; s += (y > 0.f) ? y : (__expf(y) - 1.f);
        y = a2 * bs + bsh; s += (y > 0.f) ? y : (__expf(y) - 1.f);
        y = a3 * bs + bsh; s += (y > 0.f) ? y : (__expf(y) - 1.f);
        p[((size_t)b * CH16 + m) * 1000 + t4] = s * 0.25f;
    }
}

// -------------------------------------------------------------------------
// 5) separable: depthwise K=16 pad 8 + pointwise 16x16 + BN(bn_sep) + ELU
//    + avg-pool 8 over first 1000 positions.  q: (B,16,125)
// grid: (5 w-tiles of 200, B), block 256
__global__ void k_sep(const float* __restrict__ p, const float* __restrict__ sdw,
                      const float* __restrict__ spw, const float* __restrict__ bnsep,
                      float* __restrict__ q) {
    __shared__ float ps[CH16][216];
    __shared__ float wd[256];
    __shared__ float wp[256];
    __shared__ float sc[F2d], sh[F2d];
    __shared__ float zd[CH16][200];
    __shared__ float ys[F2d][200];
    int tid = threadIdx.x;
    int wt = blockIdx.x;
    int b = blockIdx.y;
    int w0 = wt * 200;
    for (int i = tid; i < CH16 * 216; i += 256) {
        int ch = i / 216, j = i % 216;
        int wi = w0 - 8 + j;
        ps[ch][j] = (wi >= 0 && wi < 1000) ? p[((size_t)b * CH16 + ch) * 1000 + wi] : 0.f;
    }
    if (tid < 256) { wd[tid] = sdw[tid]; wp[tid] = spw[tid]; }
    if (tid < F2d) {
        float g = bnsep[0 * F2d + tid], bb = bnsep[1 * F2d + tid];
        float m = bnsep[2 * F2d + tid], v = bnsep[3 * F2d + tid];
        float s = g * rsqrtf(v + 1e-3f);
        sc[tid] = s; sh[tid] = bb - m * s;
    }
    __syncthreads();
    for (int i = tid; i < CH16 * 200; i += 256) {
        int ch = i / 200, ww = i % 200;
        float acc = 0.f;
        #pragma unroll
        for (int k = 0; k < 16; ++k) acc += wd[ch * 16 + k] * ps[ch][ww + k];
        zd[ch][ww] = acc;
    }
    __syncthreads();
    for (int i = tid; i < F2d * 200; i += 256) {
        int f2 = i / 200, ww = i % 200;
        float acc = 0.f;
        #pragma unroll
        for (int ch = 0; ch < 16; ++ch) acc += wp[f2 * 16 + ch] * zd[ch][ww];
        float y = acc * sc[f2] + sh[f2];
        ys[f2][ww] = (y > 0.f) ? y : (__expf(y) - 1.f);
    }
    __syncthreads();
    for (int i = tid; i < F2d * 25; i += 256) {
        int f2 = i / 25, jj = i % 25;
        float s = 0.f;
        #pragma unroll
        for (int k = 0; k < 8; ++k) s += ys[f2][jj * 8 + k];
        q[((size_t)b * F2d + f2) * 125 + wt * 25 + jj] = s * 0.125f;
    }
}

// -------------------------------------------------------------------------
// 6) FC: out[b,n] = q[b,:] . fc_w[n,:] + fc_b[n];  one block of 128 threads
__global__ void k_fc(const float* __restrict__ q, const float* __restrict__ fcw,
                     const float* __restrict__ fcb, float* __restrict__ out) {
    int i = threadIdx.x;               // 0..127
    int b = i >> 2, n = i & 3;
    float acc = fcb[n];
    const float* qb = q + (size_t)b * 2000;
    const float* wn = fcw + (size_t)n * 2000;
    for (int k = 0; k < 2000; ++k) acc += qb[k] * wn[k];
    out[i] = acc;
}

// -------------------------------------------------------------------------
extern "C" void kernel_launch(void* const* d_in, const int* in_sizes, int n_in,
                              void* d_out, int out_size, void* d_ws, size_t ws_size,
                              hipStream_t stream) {
    const float* x     = (const float*)d_in[0];
    const float* adj   = (const float*)d_in[1];
    const float* w1    = (const float*)d_in[2];
    const float* bnF1  = (const float*)d_in[3];
    const float* imp   = (const float*)d_in[4];
    const float* gw    = (const float*)d_in[5];
    const float* dww   = (const float*)d_in[6];
    const float* bnsp  = (const float*)d_in[7];
    const float* sdw   = (const float*)d_in[8];
    const float* spw   = (const float*)d_in[9];
    const float* bnsep = (const float*)d_in[10];
    const float* fcw   = (const float*)d_in[11];
    const float* fcb   = (const float*)d_in[12];

    const size_t HN = (size_t)Bd * F1d * Cd * Td;   // 65,536,000 floats
    float* buf0 = (float*)d_ws;
    float* buf1 = buf0 + HN;
    float* pbuf = buf1 + HN;
    float* qbuf = pbuf + (size_t)Bd * CH16 * 1000;
    float* awbuf = qbuf + (size_t)Bd * F2d * 125;   // F1*C*C floats

    dim3 gconv1(16, Cd, Bd);
    k_conv1_bn<<<gconv1, 256, 0, stream>>>(x, w1, bnF1, buf0);

    for (int l = 0; l < 3; ++l) {
        k_prep_A<<<F1d * Cd * Cd / 256, 256, 0, stream>>>(
            adj, imp + (size_t)l * F1d * Cd * Cd, awbuf);
        dim3 ge(125, Bd * F1d);
        k_einsum_wmma<<<ge, dim3(32, 2), 0, stream>>>(buf0, buf1, awbuf);
        k_gconv_bn_elu<<<gconv1, 256, 0, stream>>>(
            buf1, buf0, gw + (size_t)l * F1d * F1d * 8,
            bnF1 + (size_t)(l + 1) * 4 * F1d);
    }

    dim3 gd(4, CH16, Bd);
    k_dw_bn_elu_pool4<<<gd, 256, 0, stream>>>(buf0, dww, bnsp, pbuf);

    dim3 gs(5, Bd);
    k_sep<<<gs, 256, 0, stream>>>(pbuf, sdw, spw, bnsep, qbuf);

    k_fc<<<1, 128, 0, stream>>>(qbuf, fcw, fcb, (float*)d_out);
}